// BatchedChildSumTreeLSTM_74603581931880
// MI455X (gfx1250) — compile-verified
//
#include <hip/hip_runtime.h>

// ---------------------------------------------------------------------------
// BatchedChildSumTreeLSTM for MI455X (gfx1250), fp32 via V_WMMA_F32_16X16X4_F32
// + ASYNCcnt-tracked global->LDS staging (GLOBAL_LOAD_ASYNC_TO_LDS_B128).
// ---------------------------------------------------------------------------
#define B_      16
#define T_      256
#define T2_     16
#define H_      256
#define DEPREL_ 64
#define DEPTH_  8
#define TP2_    258          // T + 2 (padded h/c rows)
#define LDA_    260          // padded LDS stride (floats); 1040B = 65*16 -> b128-aligned rows
#define LDD_    68           // padded stride for the 16x64 deprel tile (272B = 17*16)

typedef __attribute__((ext_vector_type(2))) float v2f;
typedef __attribute__((ext_vector_type(8))) float v8f;
// exact pointee type the async-LDS builtin expects (from hipcc diagnostic):
typedef int v4i_ __attribute__((__vector_size__(4 * sizeof(int))));

#if defined(__AMDGCN__) && __has_builtin(__builtin_amdgcn_global_load_async_to_lds_b128)
#define ASYNC_LDS 1
#endif

// ---- global -> LDS staging helpers (async path when available) ------------
__device__ __forceinline__ void copy16_g2l(const float* g, float* l) {
#ifdef ASYNC_LDS
#pragma unroll
    for (int q = 0; q < 4; ++q)
        __builtin_amdgcn_global_load_async_to_lds_b128(
            (__attribute__((address_space(1))) v4i_*)(void*)(g + 4 * q),
            (__attribute__((address_space(3))) v4i_*)(void*)(l + 4 * q), 0, 0);
#else
#pragma unroll
    for (int q = 0; q < 4; ++q)
        ((float4*)l)[q] = ((const float4*)g)[q];
#endif
}
__device__ __forceinline__ void copy4_g2l(const float* g, float* l) {
#ifdef ASYNC_LDS
    __builtin_amdgcn_global_load_async_to_lds_b128(
        (__attribute__((address_space(1))) v4i_*)(void*)g,
        (__attribute__((address_space(3))) v4i_*)(void*)l, 0, 0);
#else
    *(float4*)l = *(const float4*)g;
#endif
}
__device__ __forceinline__ void async_fence() {
#ifdef ASYNC_LDS
#if __has_builtin(__builtin_amdgcn_s_wait_asynccnt)
    __builtin_amdgcn_s_wait_asynccnt(0);
#else
    asm volatile("s_wait_asynccnt 0x0" ::: "memory");
#endif
#endif
}

// ---- fp32 WMMA helpers ----------------------------------------------------
__device__ __forceinline__ v8f wmma4(v2f a, v2f b, v8f c) {
    // D = A(16x4,f32) * B(4x16,f32) + C(16x16,f32)
    return __builtin_amdgcn_wmma_f32_16x16x4_f32(false, a, false, b, (short)0, c, false, false);
}
// A fragment from LDS, row-major [16][ld]: lane lq = row M, lane-half picks K pair
__device__ __forceinline__ v2f afrag(const float* As, int ld, int k0, int hl, int lq) {
    int kr = k0 + 2 * hl;
    v2f a; a.x = As[lq * ld + kr]; a.y = As[lq * ld + kr + 1]; return a;
}
// B fragment from global, row-major [K][ldb], caller pre-offsets to column base
__device__ __forceinline__ v2f bfrag(const float* Bg, int ldb, int k0, int hl, int lq) {
    int kr = k0 + 2 * hl;
    v2f b; b.x = Bg[(size_t)kr * ldb + lq]; b.y = Bg[(size_t)(kr + 1) * ldb + lq]; return b;
}
__device__ __forceinline__ float sigm(float x) { return 1.0f / (1.0f + __expf(-x)); }

// ---------------------------------------------------------------------------
__global__ void zero_kernel(float* p, int n) {
    int i = blockIdx.x * blockDim.x + threadIdx.x;
    int stride = gridDim.x * blockDim.x;
    for (; i < n; i += stride) p[i] = 0.f;
}

// ---------------------------------------------------------------------------
// x_iouf = tok[4096,256] @ W[256,1024] + b  (hoisted, computed once)
// job = mt*64 + nt ; 8 waves/WG share the same mt (A tile in LDS)
__global__ __launch_bounds__(256) void xiouf_kernel(const float* __restrict__ tok,
                                                    const float* __restrict__ W,
                                                    const float* __restrict__ bias,
                                                    float* __restrict__ out) {
    __shared__ float sA[16 * LDA_];
    int tid = threadIdx.x;
    int wave = tid >> 5, lane = tid & 31, hl = lane >> 4, lq = lane & 15;
    int job0 = blockIdx.x * 8;
    int mt = job0 >> 6;
    {   // stage A rows mt*16..+16 (16 floats per thread, async -> LDS)
        int r = tid >> 4, seg = tid & 15;
        copy16_g2l(tok + (size_t)(mt * 16 + r) * H_ + seg * 16, &sA[r * LDA_ + seg * 16]);
    }
    async_fence();
    __syncthreads();
    int nt = (job0 + wave) & 63;
    const float* Bp = W + nt * 16;
    v8f acc = {};
    for (int k0 = 0; k0 < H_; k0 += 4)
        acc = wmma4(afrag(sA, LDA_, k0, hl, lq), bfrag(Bp, 1024, k0, hl, lq), acc);
    int n = nt * 16 + lq;
    float bv = bias[n];
#pragma unroll
    for (int i = 0; i < 8; i++) {
        int m = i + 8 * hl;
        out[(size_t)(mt * 16 + m) * 1024 + n] = acc[i] + bv;
    }
}

// ---------------------------------------------------------------------------
// One workgroup per (b,t) node. Gathers children h/c rows with max_norm=2
// renorm, computes attention pooling h_j and forget-gated child-cell sum.
__global__ __launch_bounds__(256) void child_kernel(
    const float* __restrict__ hprev, const float* __restrict__ cprev,
    const int* __restrict__ trees, const float* __restrict__ cmask,
    const float* __restrict__ deprel,
    const float* __restrict__ W_deprel, const float* __restrict__ W_hproj,
    const float* __restrict__ W_attnv, const float* __restrict__ W_hf,
    const float* __restrict__ b_hf, const float* __restrict__ xiouf,
    float* __restrict__ hj_out, float* __restrict__ fc_out) {
    __shared__ float sCH[16 * LDA_], sCC[16 * LDA_], sDep[16 * LDD_];
    __shared__ float sRH[256], sRC[256], sScH[16], sScC[16];
    __shared__ float sLog[16], sMask[16], sW[16], sFC[256];

    int tid = threadIdx.x;
    int node = blockIdx.x;
    int b = node >> 8;
    int wave = tid >> 5, lane = tid & 31, hl = lane >> 4, lq = lane & 15;

    __builtin_prefetch(W_hproj + (tid << 2), 0, 0);   // global_prefetch_b8 -> L2
    __builtin_prefetch(W_hf + (tid << 2), 0, 0);

    // ---- random-row gather: async-copy child h/c rows straight into LDS
    int j = tid >> 4, seg = tid & 15;
    int idx = trees[node * T2_ + j];
    copy16_g2l(hprev + (size_t)idx * H_ + seg * 16, &sCH[j * LDA_ + seg * 16]);
    copy16_g2l(cprev + (size_t)idx * H_ + seg * 16, &sCC[j * LDA_ + seg * 16]);
    {   // deprel tile [16][64]: 4 floats per thread
        int r = tid >> 4, c4 = (tid & 15) * 4;
        copy4_g2l(deprel + (size_t)node * T2_ * DEPREL_ + r * DEPREL_ + c4,
                  &sDep[r * LDD_ + c4]);
    }
    if (tid < 16) { sLog[tid] = 0.f; sMask[tid] = cmask[node * T2_ + tid]; }
    async_fence();
    __syncthreads();

    // ---- per-child squared norms (from LDS), then max_norm=2 renorm scale
    {
        float ssh = 0.f, ssc = 0.f;
#pragma unroll 4
        for (int k = 0; k < 16; k++) {
            float hv = sCH[j * LDA_ + seg * 16 + k]; ssh += hv * hv;
            float cv = sCC[j * LDA_ + seg * 16 + k]; ssc += cv * cv;
        }
        sRH[tid] = ssh; sRC[tid] = ssc;
    }
    __syncthreads();
    if (tid < 16) {
        float th = 0.f, tc = 0.f;
        for (int k = 0; k < 16; k++) { th += sRH[tid * 16 + k]; tc += sRC[tid * 16 + k]; }
        float nh = sqrtf(th), nc = sqrtf(tc);
        sScH[tid] = nh > 2.0f ? 2.0f / fmaxf(nh, 1e-12f) : 1.0f;
        sScC[tid] = nc > 2.0f ? 2.0f / fmaxf(nc, 1e-12f) : 1.0f;
    }
    __syncthreads();
    {
        float sh = sScH[j], sc = sScC[j];
        for (int k = 0; k < 16; k++) {
            sCH[j * LDA_ + seg * 16 + k] *= sh;
            sCC[j * LDA_ + seg * 16 + k] *= sc;
        }
    }
    __syncthreads();

    // ---- logits = tanh(ch @ W_hproj + dep @ W_deprel) @ W_attnv
    // each wave owns 32 output columns (two 16x16 WMMA tiles)
#pragma unroll
    for (int tile = 0; tile < 2; ++tile) {
        int ncol = wave * 32 + tile * 16;
        v8f acc = {};
        for (int k0 = 0; k0 < DEPREL_; k0 += 4)
            acc = wmma4(afrag(sDep, LDD_, k0, hl, lq), bfrag(W_deprel + ncol, H_, k0, hl, lq), acc);
        for (int k0 = 0; k0 < H_; k0 += 4)
            acc = wmma4(afrag(sCH, LDA_, k0, hl, lq), bfrag(W_hproj + ncol, H_, k0, hl, lq), acc);
        int n = ncol + lq;
        float wv = W_attnv[n];
#pragma unroll
        for (int i = 0; i < 8; i++) {
            int m = i + 8 * hl;
            atomicAdd(&sLog[m], tanhf(acc[i]) * wv);   // ds_add_f32
        }
    }
    __syncthreads();
    if (tid == 0) {   // masked softmax over 16 children (tiny; serial)
        float l[16], mx = -1e30f;
        for (int k = 0; k < 16; k++) {
            l[k] = sLog[k] + __logf(sMask[k] + 1e-45f);
            mx = fmaxf(mx, l[k]);
        }
        float s = 0.f;
        for (int k = 0; k < 16; k++) { l[k] = __expf(l[k] - mx); s += l[k]; }
        float inv = 1.0f / s;
        for (int k = 0; k < 16; k++) sW[k] = l[k] * inv * sMask[k];   // a * mask
    }
    __syncthreads();
    {   // h_j[n] = sum_j w_j * ch[j][n]   (one column per thread)
        float acc = 0.f;
        for (int k = 0; k < 16; k++) acc += sW[k] * sCH[k * LDA_ + tid];
        hj_out[(size_t)node * H_ + tid] = acc;
        sFC[tid] = 0.f;
    }
    __syncthreads();

    // ---- hf = ch @ W_hf + b_hf ; f = sigmoid(x_f[b, childslot, :] + hf)
    //      fc[n] = sum_j f*cc*mask (faithful x_f[:, :Tc] broadcast quirk)
#pragma unroll
    for (int tile = 0; tile < 2; ++tile) {
        int ncol = wave * 32 + tile * 16;
        v8f acc = {};
        for (int k0 = 0; k0 < H_; k0 += 4)
            acc = wmma4(afrag(sCH, LDA_, k0, hl, lq), bfrag(W_hf + ncol, H_, k0, hl, lq), acc);
        int n = ncol + lq;
        float bv = b_hf[n];
        float part = 0.f;
#pragma unroll
        for (int i = 0; i < 8; i++) {
            int m = i + 8 * hl;                                   // child slot
            float xf = xiouf[(size_t)(b * T_ + m) * 1024 + 768 + n];
            float f = sigm(xf + bv + acc[i]);
            part += f * sCC[m * LDA_ + n] * sMask[m];
        }
        atomicAdd(&sFC[n], part);
    }
    __syncthreads();
    fc_out[(size_t)node * H_ + tid] = sFC[tid];
}

// ---------------------------------------------------------------------------
// h_iou = h_j @ W_hiou + b ; gates ; c_new ; h_new -> next level buffers / out
// job = mt*16 + nt ; 8 waves/WG share mt ; i/o/u share the A fragment
__global__ __launch_bounds__(256) void gate_kernel(
    const float* __restrict__ hj, const float* __restrict__ fc,
    const float* __restrict__ xiouf, const float* __restrict__ W_hiou,
    const float* __restrict__ b_hiou,
    float* __restrict__ hnext, float* __restrict__ cnext,
    float* __restrict__ out, int last) {
    __shared__ float sA[16 * LDA_];
    int tid = threadIdx.x;
    int wave = tid >> 5, lane = tid & 31, hl = lane >> 4, lq = lane & 15;
    int job0 = blockIdx.x * 8;
    int mt = job0 >> 4;
    {
        int r = tid >> 4, seg = tid & 15;
        copy16_g2l(hj + (size_t)(mt * 16 + r) * H_ + seg * 16, &sA[r * LDA_ + seg * 16]);
    }
    async_fence();
    __syncthreads();
    int nt = (job0 + wave) & 15;
    const float* BI = W_hiou + nt * 16;
    const float* BO = W_hiou + 256 + nt * 16;
    const float* BU = W_hiou + 512 + nt * 16;
    v8f aI = {}, aO = {}, aU = {};
    for (int k0 = 0; k0 < H_; k0 += 4) {
        v2f a = afrag(sA, LDA_, k0, hl, lq);
        aI = wmma4(a, bfrag(BI, 768, k0, hl, lq), aI);
        aO = wmma4(a, bfrag(BO, 768, k0, hl, lq), aO);
        aU = wmma4(a, bfrag(BU, 768, k0, hl, lq), aU);
    }
    int n = nt * 16 + lq;
    float bI = b_hiou[n], bO = b_hiou[256 + n], bU = b_hiou[512 + n];
#pragma unroll
    for (int i = 0; i < 8; i++) {
        int m = i + 8 * hl;
        int r = mt * 16 + m;
        const float* xr = xiouf + (size_t)r * 1024;
        float ig = sigm(xr[n] + aI[i] + bI);
        float og = sigm(xr[256 + n] + aO[i] + bO);
        float ug = tanhf(xr[512 + n] + aU[i] + bU);
        float cn = ig * ug + fc[(size_t)r * H_ + n];
        float hn = og * tanhf(cn);
        if (last) {
            out[(size_t)r * H_ + n] = hn;
        } else {
            int bb = r >> 8, tt = r & 255;
            size_t o = ((size_t)bb * TP2_ + tt + 2) * H_ + n;   // +2 front pad
            hnext[o] = hn;
            cnext[o] = cn;
        }
    }
}

// ---------------------------------------------------------------------------
extern "C" void kernel_launch(void* const* d_in, const int* in_sizes, int n_in,
                              void* d_out, int out_size, void* d_ws, size_t ws_size,
                              hipStream_t stream) {
    (void)in_sizes; (void)n_in; (void)out_size; (void)ws_size;
    const float* tok      = (const float*)d_in[0];
    const int*   trees    = (const int*)d_in[1];
    const float* cmask    = (const float*)d_in[2];
    // d_in[3] = max_depth (device scalar) — compile-time 8 in the reference
    const float* dep      = (const float*)d_in[4];
    const float* W_xiouf  = (const float*)d_in[5];
    const float* b_xiouf  = (const float*)d_in[6];
    const float* W_hiou   = (const float*)d_in[7];
    const float* b_hiou   = (const float*)d_in[8];
    const float* W_hf     = (const float*)d_in[9];
    const float* b_hf     = (const float*)d_in[10];
    const float* W_deprel = (const float*)d_in[11];
    const float* W_hproj  = (const float*)d_in[12];
    const float* W_attnv  = (const float*)d_in[13];

    float* ws = (float*)d_ws;
    // Workspace layout (floats), ~42 MB total:
    const size_t off_xiouf = 0;                       // 4096 x 1024
    const size_t buf       = (size_t)B_ * TP2_ * H_;  // 1,056,768 per h/c buffer
    const size_t off_h0    = 4096ull * 1024;
    const size_t off_c0    = off_h0 + buf;
    const size_t off_h1    = off_c0 + buf;
    const size_t off_c1    = off_h1 + buf;
    const size_t off_hj    = off_c1 + buf;            // 4096 x 256
    const size_t off_fc    = off_hj + 4096ull * H_;   // 4096 x 256

    // zero h0/c0/h1/c1 (level-0 state = 0; keeps the 2 pad rows zero forever)
    zero_kernel<<<1024, 256, 0, stream>>>(ws + off_h0, (int)(4 * buf));
    // hoisted x_iouf GEMM
    xiouf_kernel<<<2048, 256, 0, stream>>>(tok, W_xiouf, b_xiouf, ws + off_xiouf);

    for (int lvl = 0; lvl < DEPTH_; ++lvl) {
        const float* hp = ws + ((lvl & 1) ? off_h1 : off_h0);
        const float* cp = ws + ((lvl & 1) ? off_c1 : off_c0);
        float* hn = ws + ((lvl & 1) ? off_h0 : off_h1);
        float* cn = ws + ((lvl & 1) ? off_c0 : off_c1);
        child_kernel<<<B_ * T_, 256, 0, stream>>>(hp, cp, trees, cmask, dep,
                                                  W_deprel, W_hproj, W_attnv, W_hf, b_hf,
                                                  ws + off_xiouf, ws + off_hj, ws + off_fc);
        gate_kernel<<<512, 256, 0, stream>>>(ws + off_hj, ws + off_fc, ws + off_xiouf,
                                             W_hiou, b_hiou, hn, cn, (float*)d_out,
                                             (lvl == DEPTH_ - 1) ? 1 : 0);
    }
}